// Model_25194278159056
// MI455X (gfx1250) — compile-verified
//
#include <hip/hip_runtime.h>

// ---------------------------------------------------------------------------
// Problem constants (from the reference)
// ---------------------------------------------------------------------------
#define B_N 32768
#define D_N 512
#define M_N 4
#define C_N 20
#define MOMF 0.9f

typedef __attribute__((ext_vector_type(2))) float v2f;
typedef __attribute__((ext_vector_type(8))) float v8f;

// Async global->LDS path (CDNA5). Guarded so both host pass and toolchains
// without the builtins still compile (synchronous float4 copy fallback).
#if defined(__gfx1250__) &&                                                   \
    __has_builtin(__builtin_amdgcn_global_load_async_to_lds_b128) &&          \
    __has_builtin(__builtin_amdgcn_s_wait_asynccnt)
#define USE_ASYNC_LDS 1
#else
#define USE_ASYNC_LDS 0
#endif

#if USE_ASYNC_LDS
typedef int vi4 __attribute__((ext_vector_type(4)));
typedef __attribute__((address_space(1))) vi4* as1_vi4p;   // global int4*
typedef __attribute__((address_space(3))) vi4* as3_vi4p;   // LDS int4*
#endif

// ---------------------------------------------------------------------------
// Workspace layout (float offsets). Regions [0, ZERO_COUNT) are zero-inited
// every call; everything after is fully written before being read.
// ---------------------------------------------------------------------------
enum : int {
  OFF_SUM_DC = 0,                              // 80 * 512
  OFF_N_DC   = 40960,                          // 80
  OFF_COVACC = 41040,                          // 4
  OFF_GRAM   = 41056,                          // 4 * 512 * 512
  ZERO_COUNT = OFF_GRAM + 4 * 512 * 512,       // zero-init prefix
  OFF_MU_DC  = ZERO_COUNT,                     // 80 * 512
  OFF_MU_D   = OFF_MU_DC + 40960,              // 4 * 512
  OFF_MU     = OFF_MU_D + 2048,                // 512
  OFF_GMEAN  = OFF_MU + 512,                   // 512
  OFF_ANCH   = OFF_GMEAN + 512,                // 80 * 512  (anchors_new)
  OFF_AGN    = OFF_ANCH + 40960,               // 20 * 512  (anchor_global_new)
  OFF_MEANP  = OFF_AGN + 10240,                // 4
  OFF_ND     = OFF_MEANP + 4                   // 4
};

// ---------------------------------------------------------------------------
// K0: zero the accumulation regions of the workspace
// ---------------------------------------------------------------------------
__global__ void k_zero(float* __restrict__ ws, int n) {
  for (int i = blockIdx.x * blockDim.x + threadIdx.x; i < n;
       i += gridDim.x * blockDim.x)
    ws[i] = 0.f;
}

// ---------------------------------------------------------------------------
// K1: fused  out = feats*(1+a[did])  +  segment sums over (domain,class)
// grid = (4 column-slices of 128, 256 row-blocks of 128 rows), block = 256
// ---------------------------------------------------------------------------
__global__ void __launch_bounds__(256)
k_proj_segsum(const float* __restrict__ feats, const int* __restrict__ labels,
              const int* __restrict__ dids, const float* __restrict__ a,
              float* __restrict__ out, float* __restrict__ ws) {
  __shared__ float s_acc[80 * 128];
  __shared__ float s_cnt[80];
  const int tid = threadIdx.x;
  const int colSlice = blockIdx.x;   // 0..3
  const int rowBlk = blockIdx.y;     // 0..255
  const int col = tid & 127;
  const int r0 = tid >> 7;           // 0..1
  const int cbase = colSlice * 128;

  for (int i = tid; i < 80 * 128; i += 256) s_acc[i] = 0.f;
  if (tid < 80) s_cnt[tid] = 0.f;
  __syncthreads();

  for (int r = r0; r < 128; r += 2) {
    const int b = rowBlk * 128 + r;
    const int did = dids[b];
    const int seg = did * C_N + labels[b];
    const float f = feats[(size_t)b * D_N + cbase + col];
    out[(size_t)b * D_N + cbase + col] =
        f * (1.f + a[did * D_N + cbase + col]);
    atomicAdd(&s_acc[seg * 128 + col], f);           // ds_add_f32
    if (col == 0 && colSlice == 0) atomicAdd(&s_cnt[seg], 1.f);
  }
  __syncthreads();

  for (int i = tid; i < 80 * 128; i += 256) {
    const int seg = i >> 7, c = i & 127;
    const float v = s_acc[i];
    if (v != 0.f) atomicAdd(&ws[OFF_SUM_DC + seg * D_N + cbase + c], v);
  }
  if (colSlice == 0 && tid < 80) {
    const float v = s_cnt[tid];
    if (v != 0.f) atomicAdd(&ws[OFF_N_DC + tid], v);
  }
}

// ---------------------------------------------------------------------------
// K2: small finalize — mu_dc, mu_d, mu, gmean_new, anchors EMA, sequential
// anchor_global EMA, reg, caa, per-domain mean part of stats loss.
// Single block, 512 threads.
// ---------------------------------------------------------------------------
__device__ __forceinline__ float pow09i(int k) {
  float p = 1.f;
  for (int i = 0; i < k; ++i) p *= MOMF;
  return p;
}

__device__ float block_reduce_512(float v, float* s_red) {
  const int tid = threadIdx.x;
  s_red[tid] = v;
  __syncthreads();
  for (int off = 256; off > 0; off >>= 1) {
    if (tid < off) s_red[tid] += s_red[tid + off];
    __syncthreads();
  }
  const float r = s_red[0];
  __syncthreads();
  return r;
}

__global__ void __launch_bounds__(512)
k_small(const float* __restrict__ a, const float* __restrict__ anchors_dc,
        const float* __restrict__ counts_dc,
        const float* __restrict__ anchor_global,
        const float* __restrict__ counts_global,
        const float* __restrict__ global_mean, float* __restrict__ out,
        float* __restrict__ ws) {
  __shared__ float s_red[512];
  __shared__ float s_ndc[80], s_p[80], s_valid[80];
  __shared__ float s_nd[4];
  __shared__ float s_kc[20], s_cgnpos[20];
  const int tid = threadIdx.x;

  if (tid < 80) {
    const float n = ws[OFF_N_DC + tid];
    s_ndc[tid] = n;
    s_p[tid] = n > 0.f ? 1.f : 0.f;
  }
  __syncthreads();
  if (tid < 20) {
    float k = 0.f, sn = 0.f;
    for (int m = 0; m < 4; ++m) {
      k += s_p[m * 20 + tid];
      sn += s_ndc[m * 20 + tid];
    }
    s_kc[tid] = k;
    s_cgnpos[tid] = (counts_global[tid] + sn) > 0.f ? 1.f : 0.f;
  }
  if (tid < 4) {
    float n = 0.f;
    for (int c = 0; c < 20; ++c) n += s_ndc[tid * 20 + c];
    s_nd[tid] = n;
    ws[OFF_ND + tid] = n;
  }
  __syncthreads();
  if (tid < 80)
    s_valid[tid] =
        ((counts_dc[tid] + s_ndc[tid]) > 0.f && s_cgnpos[tid % 20] > 0.f)
            ? 1.f
            : 0.f;
  __syncthreads();

  // mu_dc and anchors_new
  for (int idx = tid; idx < 40960; idx += 512) {
    const int seg = idx >> 9;
    const float mu = ws[OFF_SUM_DC + idx] / fmaxf(s_ndc[seg], 1.f);
    ws[OFF_MU_DC + idx] = mu;
    const float an = anchors_dc[idx];
    ws[OFF_ANCH + idx] =
        s_p[seg] > 0.f ? MOMF * an + (1.f - MOMF) * mu : an;
  }
  // mu_d
  for (int idx = tid; idx < 2048; idx += 512) {
    const int m = idx >> 9, dc = idx & 511;
    float s = 0.f;
    for (int c = 0; c < 20; ++c) s += ws[OFF_SUM_DC + (m * 20 + c) * 512 + dc];
    ws[OFF_MU_D + idx] = s / fmaxf(s_nd[m], 1.f);
  }
  // mu, gmean_new
  for (int dc = tid; dc < 512; dc += 512) {
    float t = 0.f;
    for (int seg = 0; seg < 80; ++seg) t += ws[OFF_SUM_DC + seg * 512 + dc];
    const float mu = t / (float)B_N;
    ws[OFF_MU + dc] = mu;
    ws[OFF_GMEAN + dc] = MOMF * global_mean[dc] + (1.f - MOMF) * mu;
  }
  __syncthreads();

  // anchor_global_new: 0.9^k * old + 0.1 * sum_m 0.9^suffix * mu_dc
  for (int idx = tid; idx < 10240; idx += 512) {
    const int c = idx >> 9, dc = idx & 511;
    float acc = 0.f;
    for (int m = 0; m < 4; ++m) {
      if (s_p[m * 20 + c] > 0.f) {
        int suf = 0;
        for (int m2 = m + 1; m2 < 4; ++m2) suf += (s_p[m2 * 20 + c] > 0.f);
        acc += pow09i(suf) * ws[OFF_MU_DC + (m * 20 + c) * 512 + dc];
      }
    }
    ws[OFF_AGN + idx] =
        pow09i((int)s_kc[c]) * anchor_global[idx] + (1.f - MOMF) * acc;
  }
  __syncthreads();

  // caa numerator
  float loc = 0.f;
  for (int idx = tid; idx < 40960; idx += 512) {
    const int seg = idx >> 9;
    if (s_valid[seg] > 0.f) {
      const float d =
          ws[OFF_ANCH + idx] - ws[OFF_AGN + (seg % 20) * 512 + (idx & 511)];
      loc += d * d;
    }
  }
  const float caasum = block_reduce_512(loc, s_red);

  // reg
  loc = 0.f;
  for (int idx = tid; idx < 2048; idx += 512) {
    const int m = idx >> 9;
    if (s_nd[m] > 0.f) {
      const float av = a[idx];
      const float t = av * av - 1.f;
      loc += t * t;
    }
  }
  const float regsum = block_reduce_512(loc, s_red);

  // per-domain mean part of stats loss
  for (int m = 0; m < 4; ++m) {
    loc = 0.f;
    for (int dc = tid; dc < 512; dc += 512) {
      const float d = ws[OFF_MU_D + m * 512 + dc] - ws[OFF_GMEAN + dc];
      loc += d * d;
    }
    const float s = block_reduce_512(loc, s_red);
    if (tid == 0) ws[OFF_MEANP + m] = s / 512.f;
  }

  if (tid == 0) {
    float vcnt = 0.f;
    for (int i = 0; i < 80; ++i) vcnt += s_valid[i];
    out[(size_t)B_N * D_N + 0] = regsum / 512.f / 4.f;                 // reg
    out[(size_t)B_N * D_N + 1] = (caasum / 512.f) / fmaxf(vcnt, 1.f);  // caa
  }
}

// ---------------------------------------------------------------------------
// K3: ALL FOUR per-domain Gram matrices G_m = X_m^T X_m in one pass, via
// V_WMMA_F32_16X16X4_F32. LDS slabs hold RAW feats (shared by all domains);
// the domain mask is applied only to the A fragment (product row is zeroed
// if the A row is zeroed). Double-buffered async global->LDS staging.
// Upper-triangle 64x64 tiles, split-K, global f32 atomic accumulation.
// Block = 128 threads (4 wave32s). Each wave owns a 16x64 strip x 4 domains.
// ---------------------------------------------------------------------------
#define KS 32
#define SPLITK 8
#define LDSP 72   // LDS row stride (floats): (2 rows * 72) % 64 banks = 16

__shared__ float g_lds_a[2][KS * LDSP];
__shared__ float g_lds_b[2][KS * LDSP];
__shared__ int g_dom[2][KS];

__device__ __forceinline__ void lds_copy16(float* dst, const float* src) {
#if USE_ASYNC_LDS
  __builtin_amdgcn_global_load_async_to_lds_b128((as1_vi4p)src, (as3_vi4p)dst,
                                                 0, 0);
#else
  *(float4*)dst = *(const float4*)src;
#endif
}

__device__ __forceinline__ void wait_copies_and_barrier() {
#if USE_ASYNC_LDS
  __builtin_amdgcn_s_wait_asynccnt(0);
#endif
  __syncthreads();
}

__device__ __forceinline__ void stage_copy(const float* __restrict__ feats,
                                           const int* __restrict__ dids,
                                           int buf, int kb, int i0, int j0,
                                           int tid) {
  if (tid < KS) g_dom[buf][tid] = dids[kb + tid];
#pragma unroll
  for (int it = 0; it < 4; ++it) {
    const int e = tid + it * 128;        // 0..511
    const int row = e >> 4;              // 0..31
    const int c4 = (e & 15) << 2;        // 0,4,...,60
    const float* rp = feats + (size_t)(kb + row) * D_N;
    lds_copy16(&g_lds_a[buf][row * LDSP + c4], rp + i0 + c4);
    lds_copy16(&g_lds_b[buf][row * LDSP + c4], rp + j0 + c4);
  }
}

__global__ void __launch_bounds__(128)
k_gram(const float* __restrict__ feats, const int* __restrict__ dids,
       float* __restrict__ ws) {
  const int tid = threadIdx.x;
  const int lane = tid & 31;
  const int w = tid >> 5;            // wave id 0..3

  // blockIdx.x in [0,36) -> (ti,tj) with ti<=tj
  int t = blockIdx.x, ti = 0;
  while (t >= 8 - ti) { t -= 8 - ti; ++ti; }
  const int tj = ti + t;
  const int i0 = ti * 64, j0 = tj * 64;

  const int kchunk = B_N / SPLITK;
  const int kb0 = blockIdx.y * kchunk;
  const int kb1 = kb0 + kchunk;

  v8f acc[4][4] = {};                // [domain][n-block]

  const int aoff = w * 16;
  const int mi = lane & 15;
  const int khalf = (lane < 16) ? 0 : 2;   // A/B VGPR0 holds K rows {k,k+2}

  // prologue: stage 0 into buffer 0
  stage_copy(feats, dids, 0, kb0, i0, j0, tid);
  wait_copies_and_barrier();

  int buf = 0;
  for (int kb = kb0; kb < kb1; kb += KS) {
    const int nxt = buf ^ 1;
    if (kb + KS < kb1) stage_copy(feats, dids, nxt, kb + KS, i0, j0, tid);

#pragma unroll 2
    for (int kk = 0; kk < KS; kk += 4) {
      const int kr = kk + khalf;
      const float ax = g_lds_a[buf][kr * LDSP + aoff + mi];
      const float ay = g_lds_a[buf][(kr + 1) * LDSP + aoff + mi];
      const int d0 = g_dom[buf][kr];
      const int d1 = g_dom[buf][kr + 1];
      v2f bf[4];
#pragma unroll
      for (int nb = 0; nb < 4; ++nb) {
        bf[nb].x = g_lds_b[buf][kr * LDSP + nb * 16 + mi];
        bf[nb].y = g_lds_b[buf][(kr + 1) * LDSP + nb * 16 + mi];
      }
#pragma unroll
      for (int dm = 0; dm < 4; ++dm) {
        v2f af;
        af.x = (d0 == dm) ? ax : 0.f;   // mask A only: zero row kills product
        af.y = (d1 == dm) ? ay : 0.f;
#pragma unroll
        for (int nb = 0; nb < 4; ++nb) {
          acc[dm][nb] = __builtin_amdgcn_wmma_f32_16x16x4_f32(
              false, af, false, bf[nb], (short)0, acc[dm][nb], false, false);
        }
      }
    }
    wait_copies_and_barrier();
    buf = nxt;
  }

  // write out: C/D layout -> VGPR v: lanes0-15 = row v, lanes16-31 = row v+8
  const int rowh = (lane < 16) ? 0 : 8;
  const int nj = lane & 15;
#pragma unroll
  for (int dm = 0; dm < 4; ++dm) {
    float* gram = ws + OFF_GRAM + (size_t)dm * 512 * 512;
#pragma unroll
    for (int nb = 0; nb < 4; ++nb) {
#pragma unroll
      for (int v = 0; v < 8; ++v) {
        const int gi = i0 + w * 16 + rowh + v;
        const int gj = j0 + nb * 16 + nj;
        atomicAdd(&gram[gi * 512 + gj], acc[dm][nb][v]);
      }
    }
  }
}

// ---------------------------------------------------------------------------
// K4: covariance finalize.  cov = (G_tot - B*mu*mu^T)/(B+eps);
// cov_d[m] = (G_m - n_m*mu_m*mu_m^T)/(n_m+eps);  accumulate sum (cov_d-gcov)^2.
// ---------------------------------------------------------------------------
__global__ void __launch_bounds__(256)
k_stats_cov(const float* __restrict__ global_cov, float* __restrict__ ws) {
  __shared__ float s_red[256];
  const int idx = blockIdx.x * 256 + threadIdx.x;   // 0..262143
  const int i = idx >> 9, j = idx & 511;
  const int lin = (i <= j) ? (i * 512 + j) : (j * 512 + i);  // symmetry mirror

  const float g0 = ws[OFF_GRAM + 0 * 262144 + lin];
  const float g1 = ws[OFF_GRAM + 1 * 262144 + lin];
  const float g2 = ws[OFF_GRAM + 2 * 262144 + lin];
  const float g3 = ws[OFF_GRAM + 3 * 262144 + lin];
  const float gt = g0 + g1 + g2 + g3;
  const float mui = ws[OFF_MU + i], muj = ws[OFF_MU + j];
  const float cov = (gt - (float)B_N * mui * muj) / ((float)B_N + 1e-6f);
  const float gcov = MOMF * global_cov[idx] + (1.f - MOMF) * cov;

  const float gm[4] = {g0, g1, g2, g3};
  float part[4];
#pragma unroll
  for (int m = 0; m < 4; ++m) {
    const float n = ws[OFF_ND + m];
    const float covd =
        (gm[m] - n * ws[OFF_MU_D + m * 512 + i] * ws[OFF_MU_D + m * 512 + j]) /
        (n + 1e-6f);
    const float d = covd - gcov;
    part[m] = d * d;
  }
  for (int m = 0; m < 4; ++m) {
    s_red[threadIdx.x] = part[m];
    __syncthreads();
    for (int off = 128; off > 0; off >>= 1) {
      if (threadIdx.x < off) s_red[threadIdx.x] += s_red[threadIdx.x + off];
      __syncthreads();
    }
    if (threadIdx.x == 0) atomicAdd(&ws[OFF_COVACC + m], s_red[0]);
    __syncthreads();
  }
}

// ---------------------------------------------------------------------------
// K5: final stats scalar
// ---------------------------------------------------------------------------
__global__ void k_final(float* __restrict__ out, const float* __restrict__ ws) {
  if (threadIdx.x == 0 && blockIdx.x == 0) {
    float P = 0.f, s = 0.f;
    for (int m = 0; m < 4; ++m) {
      const float n = ws[OFF_ND + m];
      if (n > 0.f) {
        P += 1.f;
        s += ws[OFF_MEANP + m] + ws[OFF_COVACC + m] / 262144.f;
      }
    }
    out[(size_t)B_N * D_N + 2] = s / fmaxf(P, 1.f);
  }
}

// ---------------------------------------------------------------------------
extern "C" void kernel_launch(void* const* d_in, const int* in_sizes, int n_in,
                              void* d_out, int out_size, void* d_ws,
                              size_t ws_size, hipStream_t stream) {
  const float* feats = (const float*)d_in[0];
  const int* labels = (const int*)d_in[1];
  const int* dids = (const int*)d_in[2];
  const float* a = (const float*)d_in[3];
  const float* anchors_dc = (const float*)d_in[4];
  const float* counts_dc = (const float*)d_in[5];
  const float* anchor_global = (const float*)d_in[6];
  const float* counts_global = (const float*)d_in[7];
  const float* global_mean = (const float*)d_in[8];
  const float* global_cov = (const float*)d_in[9];
  float* out = (float*)d_out;
  float* ws = (float*)d_ws;

  k_zero<<<2048, 256, 0, stream>>>(ws, (int)ZERO_COUNT);
  k_proj_segsum<<<dim3(4, 256), 256, 0, stream>>>(feats, labels, dids, a, out,
                                                  ws);
  k_small<<<1, 512, 0, stream>>>(a, anchors_dc, counts_dc, anchor_global,
                                 counts_global, global_mean, out, ws);
  k_gram<<<dim3(36, SPLITK), 128, 0, stream>>>(feats, dids, ws);
  k_stats_cov<<<1024, 256, 0, stream>>>(global_cov, ws);
  k_final<<<1, 32, 0, stream>>>(out, ws);
}